// SiSoMmsePicDetector_79559974191758
// MI455X (gfx1250) — compile-verified
//
#include <hip/hip_runtime.h>
#include <hip/hip_bf16.h>
#include <math.h>

// ---------------------------------------------------------------------------
// SiSo MMSE-PIC detector, CDNA5 (gfx1250), wave32.
// One wave32 workgroup per batch item (B = 32768). M = 2K = 32 = wave width.
// WMMA f32 16x16x4 used for the Gram products Hr^T Hr, Hi^T Hi, Hr^T Hi.
// ---------------------------------------------------------------------------

#define BB_   32768
#define MM    32      // rx antennas
#define KK    16      // users
#define K2    32      // 2K
#define NBITS 4
#define NPTS  16
#define EPS_F 1.0e-4f

typedef float v2f __attribute__((ext_vector_type(2)));
typedef float v8f __attribute__((ext_vector_type(8)));

// 16-QAM Gray constellation: Re from bits (b0,b2), Im from (b1,b3);
// pam(x,y) = (1-2x)(1+2y), normalized by 1/sqrt(10).
__device__ __forceinline__ float pointRe(int p) {
  int x = (p >> 3) & 1, y = (p >> 1) & 1;
  return (float)((1 - 2 * x) * (1 + 2 * y)) * 0.31622776601683794f;
}
__device__ __forceinline__ float pointIm(int p) {
  int x = (p >> 2) & 1, y = p & 1;
  return (float)((1 - 2 * x) * (1 + 2 * y)) * 0.31622776601683794f;
}
// log_sigmoid(x) = min(x,0) - log1p(exp(-|x|))
__device__ __forceinline__ float log_sigmoid(float x) {
  return fminf(x, 0.0f) - log1pf(expf(-fabsf(x)));
}

struct __align__(16) Smem {
  // Cholesky L (complex, lower) lives only until the solves are done; the
  // Gauss-Jordan augmented matrix + Gri transpose bounce reuse the space.
  union {
    struct { float re[MM][MM + 1]; float im[MM][MM + 1]; } L;          // 8448 B
    struct { float aug[K2][2 * K2 + 2]; float gtmp[KK][KK + 1]; } inv; // 9536 B
  } u;
  float Hr[MM][KK + 1];   // hw real (in: h real, solved in place)
  float Hi[MM][KK + 1];   // hw imag
  float gr[K2][K2 + 1];   // real-embedded Gram matrix (symmetric)
  float ywr[MM], ywi[MM]; // yw (in: y, solved in place)
  float ymfr[KK], ymfi[KK];
  float xr[KK], xi[KK];   // x_hat
  float var[KK];          // var_x
  float cr[KK], ci[KK];   // y_mf - g @ x_hat
  float logp[KK][NPTS + 1];
  float xh[K2];
  float noeff[KK];
  float llra[KK][NBITS];
  float llrd[KK][NBITS];
};

__global__ __launch_bounds__(32) void siso_mmse_pic_kernel(
    const float* __restrict__ y,      // (B, M) complex64 interleaved
    const float* __restrict__ h,      // (B, M, K) complex64 interleaved
    const float* __restrict__ prior,  // (B, K, NB) f32
    const float* __restrict__ s,      // (B, M, M) complex64 interleaved
    float* __restrict__ out)          // (B, K, NB) f32
{
  __shared__ Smem sm;
  const int b    = blockIdx.x;
  const int lane = threadIdx.x;      // 0..31, exactly one wave32
  const int lo   = lane & 15;
  const int hi   = lane >> 4;

  // ---------------- load: s -> L, h -> Hr/Hi, y -> yw, prior -> llrd -------
  {
    const float* sb = s + (size_t)b * MM * MM * 2;
    for (int idx = lane; idx < MM * MM; idx += 32) {
      int i = idx >> 5, j = idx & 31;
      sm.u.L.re[i][j] = sb[2 * idx];
      sm.u.L.im[i][j] = sb[2 * idx + 1];
    }
    const float* hb = h + (size_t)b * MM * KK * 2;
    for (int idx = lane; idx < MM * KK; idx += 32) {
      int m = idx >> 4, k = idx & 15;
      sm.Hr[m][k] = hb[2 * idx];
      sm.Hi[m][k] = hb[2 * idx + 1];
    }
    const float* yb = y + (size_t)b * MM * 2;
    sm.ywr[lane] = yb[2 * lane];
    sm.ywi[lane] = yb[2 * lane + 1];
    const float* pb = prior + (size_t)b * KK * NBITS;
    for (int idx = lane; idx < KK * NBITS; idx += 32)
      sm.llrd[idx >> 2][idx & 3] = pb[idx];
  }
  __syncthreads();

  // ---------------- complex Cholesky (lower), lane-per-row -----------------
  for (int j = 0; j < MM; ++j) {
    float d = sm.u.L.re[j][j];                 // broadcast LDS reads
    for (int p = 0; p < j; ++p) {
      float lr = sm.u.L.re[j][p], li = sm.u.L.im[j][p];
      d -= lr * lr + li * li;
    }
    d = sqrtf(fmaxf(d, 1.0e-20f));
    float invd = 1.0f / d;
    if (lane == j) { sm.u.L.re[j][j] = d; sm.u.L.im[j][j] = 0.0f; }
    if (lane > j) {
      float ar = sm.u.L.re[lane][j], ai = sm.u.L.im[lane][j];
      for (int p = 0; p < j; ++p) {
        float lr = sm.u.L.re[lane][p], li = sm.u.L.im[lane][p];
        float mr = sm.u.L.re[j][p],    mi = sm.u.L.im[j][p];
        ar -= lr * mr + li * mi;                // a -= l * conj(m)
        ai -= li * mr - lr * mi;
      }
      sm.u.L.re[lane][j] = ar * invd;
      sm.u.L.im[lane][j] = ai * invd;
    }
    __syncthreads();
  }

  // ------- forward substitution, in place: [hw | yw] = L^-1 [h | y] --------
  // lane c<16 -> column c of hw; lane 16 -> yw; lanes 17..31 idle.
  for (int m = 0; m < MM; ++m) {
    if (lane < KK + 1) {
      float xr_, xi_;
      if (lane < KK) { xr_ = sm.Hr[m][lane]; xi_ = sm.Hi[m][lane]; }
      else           { xr_ = sm.ywr[m];      xi_ = sm.ywi[m];      }
      for (int p = 0; p < m; ++p) {
        float lr = sm.u.L.re[m][p], li = sm.u.L.im[m][p];
        float vr, vi;
        if (lane < KK) { vr = sm.Hr[p][lane]; vi = sm.Hi[p][lane]; }
        else           { vr = sm.ywr[p];      vi = sm.ywi[p];      }
        xr_ -= lr * vr - li * vi;
        xi_ -= lr * vi + li * vr;
      }
      float invd = 1.0f / sm.u.L.re[m][m];
      xr_ *= invd; xi_ *= invd;
      if (lane < KK) { sm.Hr[m][lane] = xr_; sm.Hi[m][lane] = xi_; }
      else           { sm.ywr[m] = xr_;      sm.ywi[m] = xi_;      }
    }
    __syncthreads();
  }

  // ---------------- y_mf = hw^H yw ----------------------------------------
  if (lane < KK) {
    float ar = 0.0f, ai = 0.0f;
    for (int m = 0; m < MM; ++m) {
      float hr_ = sm.Hr[m][lane], hi_ = sm.Hi[m][lane];
      float yr_ = sm.ywr[m],      yi_ = sm.ywi[m];
      ar += hr_ * yr_ + hi_ * yi_;              // conj(h) * y
      ai += hr_ * yi_ - hi_ * yr_;
    }
    sm.ymfr[lane] = ar; sm.ymfi[lane] = ai;
  }
  __syncthreads();

  // ---------------- g via WMMA: Grr=Hr^T Hr, Gii=Hi^T Hi, Gri=Hr^T Hi -----
  // A(16x4) lane layout: m = lane%16, k = v + 2*(lane/16)  (ISA 7.12.2)
  // B(4x16)  lane layout: n = lane%16, k = v + 2*(lane/16)
  // => A-regs of H^T equal B-regs of H: one LDS read feeds both operands.
  v8f grr = {}; v8f gii = {}; v8f gri = {};
  for (int kc = 0; kc < 8; ++kc) {
    int r0 = kc * 4 + 2 * hi;
    v2f ar, ai;
    ar.x = sm.Hr[r0][lo];     ar.y = sm.Hr[r0 + 1][lo];
    ai.x = sm.Hi[r0][lo];     ai.y = sm.Hi[r0 + 1][lo];
    grr = __builtin_amdgcn_wmma_f32_16x16x4_f32(false, ar, false, ar,
                                                (short)0, grr, false, false);
    gii = __builtin_amdgcn_wmma_f32_16x16x4_f32(false, ai, false, ai,
                                                (short)0, gii, false, false);
    gri = __builtin_amdgcn_wmma_f32_16x16x4_f32(false, ar, false, ai,
                                                (short)0, gri, false, false);
  }
  // Bounce Gri through LDS to read its transpose (Im g = Gri - Gri^T).
  #pragma unroll
  for (int r = 0; r < 8; ++r)
    sm.u.inv.gtmp[r + 8 * hi][lo] = gri[r];     // C layout: row r+8*hi, col lo
  __syncthreads();
  #pragma unroll
  for (int r = 0; r < 8; ++r) {
    int i = r + 8 * hi, j = lo;
    float reg = grr[r] + gii[r];                // Re g[i][j]
    float img = gri[r] - sm.u.inv.gtmp[j][i];   // Im g[i][j]
    sm.gr[i][j]           = reg;
    sm.gr[i + 16][j + 16] = reg;
    sm.gr[i + 16][j]      = img;
    sm.gr[i][j + 16]      = -img;
  }
  __syncthreads();

  // ---------------- 2 MMSE-PIC iterations ---------------------------------
  for (int it = 0; it < 2; ++it) {
    for (int idx = lane; idx < KK * NBITS; idx += 32)
      sm.llra[idx >> 2][idx & 3] = sm.llrd[idx >> 2][idx & 3];
    __syncthreads();

    // logp[k][p] = sum_b log_sigmoid(alab[p][b] * llr_a[k][b]); 256 pairs / wave
    {
      int p = lo;
      for (int kk2 = 0; kk2 < 8; ++kk2) {
        int k = 8 * hi + kk2;
        float acc = 0.0f;
        #pragma unroll
        for (int bb = 0; bb < NBITS; ++bb) {
          float sgn = ((p >> (3 - bb)) & 1) ? 1.0f : -1.0f;
          acc += log_sigmoid(sgn * sm.llra[k][bb]);
        }
        sm.logp[k][p] = acc;
      }
    }
    __syncthreads();

    // softmax over points -> x_hat, var_x
    if (lane < KK) {
      float mx = -1.0e30f;
      for (int p = 0; p < NPTS; ++p) mx = fmaxf(mx, sm.logp[lane][p]);
      float Z = 0.0f, sxr = 0.0f, sxi = 0.0f, se2 = 0.0f;
      for (int p = 0; p < NPTS; ++p) {
        float e  = expf(sm.logp[lane][p] - mx);
        float pr = pointRe(p), pi_ = pointIm(p);
        Z += e; sxr += e * pr; sxi += e * pi_;
        se2 += e * (pr * pr + pi_ * pi_);
      }
      float invZ = 1.0f / Z;
      float xr_ = sxr * invZ, xi_ = sxi * invZ;
      sm.xr[lane] = xr_; sm.xi[lane] = xi_;
      sm.var[lane] = se2 * invZ - (xr_ * xr_ + xi_ * xi_);
    }
    __syncthreads();

    // c = y_mf - g @ x_hat   (g re/im are sub-blocks of gr)
    if (lane < KK) {
      float gxr = 0.0f, gxi = 0.0f;
      for (int j = 0; j < KK; ++j) {
        float gre = sm.gr[lane][j], gim = sm.gr[lane + 16][j];
        float xr_ = sm.xr[j],       xi_ = sm.xi[j];
        gxr += gre * xr_ - gim * xi_;
        gxi += gre * xi_ + gim * xr_;
      }
      sm.cr[lane] = sm.ymfr[lane] - gxr;
      sm.ci[lane] = sm.ymfi[lane] - gxi;
    }
    __syncthreads();

    // Aug = [ gr*diag(v2) + I | I ], lane-per-row
    for (int j = 0; j < K2; ++j) {
      float v = sm.var[j & 15];
      sm.u.inv.aug[lane][j]      = sm.gr[lane][j] * v + (lane == j ? 1.0f : 0.0f);
      sm.u.inv.aug[lane][K2 + j] = (lane == j) ? 1.0f : 0.0f;
    }
    __syncthreads();

    // Gauss-Jordan (no pivoting: A = I + Gr*D, Gr PSD => diag stays >= ~1)
    for (int k = 0; k < K2; ++k) {
      if (lane == k) {
        float invp = 1.0f / sm.u.inv.aug[k][k];
        for (int j = 0; j < 2 * K2; ++j) sm.u.inv.aug[k][j] *= invp;
      }
      __syncthreads();
      if (lane != k) {
        float f = sm.u.inv.aug[lane][k];
        for (int j = 0; j < 2 * K2; ++j)
          sm.u.inv.aug[lane][j] -= f * sm.u.inv.aug[k][j];
      }
      __syncthreads();
    }

    // mu = diag(A^-1 gr) (gr symmetric); xh = (A^-1 ytr_row) / mu; rho
    {
      int r = lane & 15;
      float mu = 0.0f, num = 0.0f;
      float crr = sm.cr[r], cii = sm.ci[r];
      for (int j = 0; j < K2; ++j) {
        float a = sm.u.inv.aug[lane][K2 + j];
        mu += a * sm.gr[lane][j];
        int jc = j & 15;
        float gre = sm.gr[r][jc], gim = sm.gr[r + 16][jc];
        float xr_ = sm.xr[jc],    xi_ = sm.xi[jc];
        float w = (j < KK) ? (crr + gre * xr_ - gim * xi_)   // Re(y_mf_pic[j][r])
                           : (cii + gre * xi_ + gim * xr_);  // Im(y_mf_pic[j][r])
        num += a * w;
      }
      sm.xh[lane] = num / mu;
      if (lane < KK) {
        float rho = mu / fmaxf(1.0f - sm.var[r] * mu, EPS_F);
        sm.noeff[lane] = 1.0f / rho;
      }
    }
    __syncthreads();

    // max-log LLRs
    if (lane < KK) {
      float xer = sm.xh[lane], xei = sm.xh[lane + 16];
      float inv_no = 1.0f / sm.noeff[lane];
      float mx1[NBITS], mx0[NBITS];
      #pragma unroll
      for (int bb = 0; bb < NBITS; ++bb) { mx1[bb] = -1.0e30f; mx0[bb] = -1.0e30f; }
      for (int p = 0; p < NPTS; ++p) {
        float dr = xer - pointRe(p), di = xei - pointIm(p);
        float e  = -(dr * dr + di * di) * inv_no + sm.logp[lane][p];
        #pragma unroll
        for (int bb = 0; bb < NBITS; ++bb) {
          if ((p >> (3 - bb)) & 1) mx1[bb] = fmaxf(mx1[bb], e);
          else                     mx0[bb] = fmaxf(mx0[bb], e);
        }
      }
      #pragma unroll
      for (int bb = 0; bb < NBITS; ++bb)
        sm.llrd[lane][bb] = mx1[bb] - mx0[bb];
    }
    __syncthreads();
  }

  // ---------------- out = llr_d - llr_a -----------------------------------
  float* ob = out + (size_t)b * KK * NBITS;
  for (int idx = lane; idx < KK * NBITS; idx += 32)
    ob[idx] = sm.llrd[idx >> 2][idx & 3] - sm.llra[idx >> 2][idx & 3];
}

extern "C" void kernel_launch(void* const* d_in, const int* in_sizes, int n_in,
                              void* d_out, int out_size, void* d_ws, size_t ws_size,
                              hipStream_t stream) {
  (void)in_sizes; (void)n_in; (void)d_ws; (void)ws_size;
  const float* y     = (const float*)d_in[0];
  const float* h     = (const float*)d_in[1];
  const float* prior = (const float*)d_in[2];
  const float* s     = (const float*)d_in[3];
  float* out         = (float*)d_out;
  const int B = out_size / (KK * NBITS);       // 32768
  hipLaunchKernelGGL(siso_mmse_pic_kernel, dim3(B), dim3(32), 0, stream,
                     y, h, prior, s, out);
}